// SparseGCNLayer_27487790695251
// MI455X (gfx1250) — compile-verified
//
#include <hip/hip_runtime.h>
#include <hip/hip_bf16.h>

#define N_NODES 100000
#define N_EDGES 3200000
#define D_IN    256
#define D_OUT   256

typedef __attribute__((ext_vector_type(2))) float v2f;
typedef __attribute__((ext_vector_type(8))) float v8f;

// ---------------------------------------------------------------------------
// Workspace layout (bytes):
//   agg    : N_NODES * D_IN * 4      = 102,400,000
//   counts : N_NODES * 4             = 400,000
//   starts : N_NODES * 4             = 400,000
//   cursor : N_NODES * 4             = 400,000
//   perm   : N_EDGES * 4             = 12,800,000
// total ~ 116.4 MB
// ---------------------------------------------------------------------------

__global__ void k_zero_f4(float4* p, int n4) {
    int i = blockIdx.x * blockDim.x + threadIdx.x;
    if (i < n4) p[i] = make_float4(0.f, 0.f, 0.f, 0.f);
}

__global__ void k_zero_i32(int* p, int n) {
    int i = blockIdx.x * blockDim.x + threadIdx.x;
    if (i < n) p[i] = 0;
}

__global__ void k_count(const int* __restrict__ row, int* __restrict__ counts, int E) {
    int e = blockIdx.x * blockDim.x + threadIdx.x;
    if (e < E) atomicAdd(&counts[row[e]], 1);
}

// Single-block exclusive scan over counts -> starts (and cursor copy).
__global__ void k_scan(const int* __restrict__ counts,
                       int* __restrict__ starts,
                       int* __restrict__ cursor, int n) {
    __shared__ int sdata[1024];
    __shared__ int carry;
    const int tid = threadIdx.x;
    if (tid == 0) carry = 0;
    __syncthreads();
    for (int base = 0; base < n; base += 1024) {
        int i = base + tid;
        int v = (i < n) ? counts[i] : 0;
        sdata[tid] = v;
        __syncthreads();
        // Hillis-Steele inclusive scan
        for (int off = 1; off < 1024; off <<= 1) {
            int t = (tid >= off) ? sdata[tid - off] : 0;
            __syncthreads();
            sdata[tid] += t;
            __syncthreads();
        }
        int excl = sdata[tid] - v;
        if (i < n) {
            int s = carry + excl;
            starts[i] = s;
            cursor[i] = s;
        }
        int total = sdata[1023];
        __syncthreads();
        if (tid == 0) carry += total;
        __syncthreads();
    }
}

__global__ void k_scatter(const int* __restrict__ row,
                          int* __restrict__ cursor,
                          int* __restrict__ perm, int E) {
    int e = blockIdx.x * blockDim.x + threadIdx.x;
    if (e < E) {
        int r = row[e];
        int pos = atomicAdd(&cursor[r], 1);
        perm[pos] = e;
    }
}

// One block (256 threads) per node; thread t owns channel t.
// Edge metadata staged through LDS; accumulation fully in registers.
__global__ void __launch_bounds__(256)
k_spmm(const float* __restrict__ x,
       const int* __restrict__ col,
       const float* __restrict__ val,
       const int* __restrict__ perm,
       const int* __restrict__ starts,
       const int* __restrict__ counts,
       float* __restrict__ agg) {
    __shared__ int   s_col[64];
    __shared__ float s_val[64];
    const int node = blockIdx.x;
    const int t    = threadIdx.x;              // channel 0..255
    const int s    = starts[node];
    const int cnt  = counts[node];
    float acc = 0.f;
    for (int base = 0; base < cnt; base += 64) {
        int chunk = min(64, cnt - base);
        if (t < chunk) {
            int eid  = perm[s + base + t];
            s_col[t] = col[eid];
            s_val[t] = val[eid];
        }
        __syncthreads();
        #pragma unroll 4
        for (int j = 0; j < chunk; ++j) {
            acc += s_val[j] * x[(size_t)s_col[j] * D_IN + t];
        }
        __syncthreads();
    }
    agg[(size_t)node * D_IN + t] = acc;
}

// out[100000,256] = agg[100000,256] @ W^T[256,256] + b
// One wave per 16x16 output tile; K swept in steps of 4 via f32 WMMA.
__global__ void __launch_bounds__(256)
k_gemm_wmma(const float* __restrict__ agg,
            const float* __restrict__ W,      // [D_OUT, D_IN] row-major
            const float* __restrict__ bias,
            float* __restrict__ out) {
    const int wid   = (blockIdx.x * blockDim.x + threadIdx.x) >> 5;
    const int tileN = wid & 15;                 // 16 N-tiles (D_OUT/16)
    const int tileM = wid >> 4;                 // 6250 M-tiles
    if (tileM >= (N_NODES / 16)) return;        // uniform per wave

    const int lane = threadIdx.x & 31;
    const int half = lane >> 4;                 // 0: K+{0,1}, 1: K+{2,3}
    const int l16  = lane & 15;

    // A (16x4 f32): lane l16 supplies row (tileM*16+l16), two consecutive K.
    const float* arow = agg + (size_t)(tileM * 16 + l16) * D_IN + 2 * half;
    // B (4x16 f32): B[k][n] = W[n][k]; lane l16 supplies column n=l16.
    const float* brow = W   + (size_t)(tileN * 16 + l16) * D_IN + 2 * half;

    v8f c = {};
    #pragma unroll 8
    for (int k0 = 0; k0 < D_IN; k0 += 4) {
        v2f a = *(const v2f*)(arow + k0);
        v2f b = *(const v2f*)(brow + k0);
        // (neg_a, A, neg_b, B, c_mod, C, reuse_a, reuse_b)
        c = __builtin_amdgcn_wmma_f32_16x16x4_f32(false, a, false, b,
                                                  (short)0, c, false, false);
    }

    const float bv = bias[tileN * 16 + l16];
    // C/D layout: VGPR j -> M = j + 8*half, N = l16.
    float* obase = out + (size_t)(tileM * 16 + 8 * half) * D_OUT + tileN * 16 + l16;
    #pragma unroll
    for (int j = 0; j < 8; ++j) {
        obase[(size_t)j * D_OUT] = c[j] + bv;
    }
}

extern "C" void kernel_launch(void* const* d_in, const int* in_sizes, int n_in,
                              void* d_out, int out_size, void* d_ws, size_t ws_size,
                              hipStream_t stream) {
    const float* x        = (const float*)d_in[0];
    const int*   edge_row = (const int*)  d_in[1];
    const int*   edge_col = (const int*)  d_in[2];
    const float* edge_val = (const float*)d_in[3];
    const float* W        = (const float*)d_in[4];
    const float* b        = (const float*)d_in[5];
    float* out = (float*)d_out;

    char* ws = (char*)d_ws;
    float* agg   = (float*)ws;                 ws += (size_t)N_NODES * D_IN * sizeof(float);
    int*  counts = (int*)ws;                   ws += (size_t)N_NODES * sizeof(int);
    int*  starts = (int*)ws;                   ws += (size_t)N_NODES * sizeof(int);
    int*  cursor = (int*)ws;                   ws += (size_t)N_NODES * sizeof(int);
    int*  perm   = (int*)ws;

    // 1) zero agg + counts
    {
        int n4 = (N_NODES * D_IN) / 4;                       // 6,400,000
        k_zero_f4<<<(n4 + 255) / 256, 256, 0, stream>>>((float4*)agg, n4);
        k_zero_i32<<<(N_NODES + 255) / 256, 256, 0, stream>>>(counts, N_NODES);
    }
    // 2) per-row edge histogram
    k_count<<<(N_EDGES + 255) / 256, 256, 0, stream>>>(edge_row, counts, N_EDGES);
    // 3) exclusive scan -> starts, cursor
    k_scan<<<1, 1024, 0, stream>>>(counts, starts, cursor, N_NODES);
    // 4) scatter edge ids into CSR order
    k_scatter<<<(N_EDGES + 255) / 256, 256, 0, stream>>>(edge_row, cursor, perm, N_EDGES);
    // 5) conflict-free SpMM aggregation (x is L2-resident: 102.4MB < 192MB L2)
    k_spmm<<<N_NODES, 256, 0, stream>>>(x, edge_col, edge_val, perm, starts, counts, agg);
    // 6) dense GEMM + bias via f32 WMMA: 100000 tiles = 12500 blocks x 8 waves
    {
        int totalTiles = (N_NODES / 16) * (D_OUT / 16);      // 100000
        int blocks = (totalTiles * 32 + 255) / 256;          // 12500
        k_gemm_wmma<<<blocks, 256, 0, stream>>>(agg, W, b, out);
    }
}